// SyncSwitchWhiten2d_25769804240
// MI455X (gfx1250) — compile-verified
//
#include <hip/hip_runtime.h>
#include <hip/hip_bf16.h>

// ---- problem constants (from reference) ----
#define HWSZ   4096          // H*W
#define NB     32            // N
#define CCH    256           // C
#define GRP    16            // groups
#define CPG    16            // channels per group
#define NHWF   131072.0f     // N*H*W
#define SSTR   288           // per-instance stats stride in floats (256 Q + 16 S, padded)
#define EPSV   1e-5f

typedef float v2f __attribute__((ext_vector_type(2)));
typedef float v8f __attribute__((ext_vector_type(8)));

// D(16x16 f32) = A(16x4 f32) * B(4x16 f32) + C
static __device__ __forceinline__ v8f wmma4(v2f a, v2f b, v8f c) {
  // 8 args: (neg_a, A, neg_b, B, c_mod, C, reuse_a, reuse_b)
  return __builtin_amdgcn_wmma_f32_16x16x4_f32(false, a, false, b, (short)0, c,
                                               false, false);
}

// 16x16x16 matmul from two row-major 16x16 LDS tiles, via 4 chained WMMAs.
// A-frag lane: m = lane16, K in {4t+kh, 4t+kh+1}; B-frag lane: n = lane16, same K split.
static __device__ __forceinline__ v8f mul16(const float* As, const float* Bs,
                                            int lane16, int kh) {
  v8f acc = {};
#pragma unroll
  for (int t = 0; t < 4; ++t) {
    int k = 4 * t + kh;
    v2f a, b;
    a.x = As[lane16 * 16 + k];
    a.y = As[lane16 * 16 + k + 1];
    b.x = Bs[k * 16 + lane16];
    b.y = Bs[(k + 1) * 16 + lane16];
    acc = wmma4(a, b, acc);
  }
  return acc;
}

// Store C/D-layout v8f into a row-major 16x16 LDS tile.
static __device__ __forceinline__ void st16(float* Ds, int lane16, int half,
                                            const v8f& d) {
#pragma unroll
  for (int v = 0; v < 8; ++v) Ds[(v + half * 8) * 16 + lane16] = d[v];
}

// ---------------------------------------------------------------------------
// K1: per-instance sufficient statistics. One block (8 waves) per (n, group).
// Q[16][16] = X * X^T via v_wmma_f32_16x16x4_f32 (A-frag == B-frag since B=A^T),
// S[16] = per-channel sums. Reduced across waves with ds_add_f32.
// Regular (RT) loads intentionally populate L2: x (128 MB) fits in the 192 MB
// global L2 and is re-read by K3.
// ---------------------------------------------------------------------------
__global__ void k_stats(const float* __restrict__ x, float* __restrict__ statsA) {
  const int inst = blockIdx.x;               // n*16 + gi
  const int n = inst >> 4, gi = inst & 15, cb = gi * CPG;
  const int tid = threadIdx.x;
  const int lane = tid & 31, wave = tid >> 5;
  const int lane16 = lane & 15, half = lane >> 4, kh = half * 2;

  __shared__ float sQ[256];
  __shared__ float sS[16];
  if (tid < 256) sQ[tid] = 0.f;
  if (tid < 16) sS[tid] = 0.f;
  __syncthreads();

  // this lane streams channel (cb+lane16); each wave owns 512 spatial positions
  const float* row = x + (size_t)(n * CCH + cb + lane16) * HWSZ;
  const int kbase = wave * 512 + kh;

  v8f acc = {};
  float ssum = 0.f;
#pragma unroll 4
  for (int i = 0; i < 128; ++i) {
    v2f v = *(const v2f*)(row + kbase + 4 * i);   // global_load_b64
    ssum += v.x + v.y;
    acc = wmma4(v, v, acc);                       // Q += X_chunk * X_chunk^T
  }

  atomicAdd(&sS[lane16], ssum);                   // ds_add_f32
#pragma unroll
  for (int v = 0; v < 8; ++v) {
    int r = v + half * 8;                         // C/D layout row
    atomicAdd(&sQ[r * 16 + lane16], acc[v]);
  }
  __syncthreads();

  float* dst = statsA + (size_t)inst * SSTR;
  if (tid < 256) dst[tid] = sQ[tid];
  if (tid < 16) dst[256 + tid] = sS[tid];
}

// ---------------------------------------------------------------------------
// K1.5: aggregate per-instance stats over the batch -> per-group (BN) stats.
// ---------------------------------------------------------------------------
__global__ void k_bnagg(const float* __restrict__ statsA, float* __restrict__ statsB) {
  const int gi = blockIdx.x, tid = threadIdx.x;
  float acc = 0.f;
  for (int n = 0; n < NB; ++n) acc += statsA[(size_t)(n * GRP + gi) * SSTR + tid];
  statsB[(size_t)gi * SSTR + tid] = acc;
  if (tid < 16) {
    float a2 = 0.f;
    for (int n = 0; n < NB; ++n)
      a2 += statsA[(size_t)(n * GRP + gi) * SSTR + 256 + tid];
    statsB[(size_t)gi * SSTR + 256 + tid] = a2;
  }
}

// ---------------------------------------------------------------------------
// K2: per-instance mixed stats + Newton-Schulz (all 16x16 matmuls on WMMA).
// One wave per instance. Emits folded wm2 = diag(weight)*wm*sqrt(rTr) and
// wbias = bias - wm2 @ mean so K3 is GEMM + bias.
// ---------------------------------------------------------------------------
__global__ void __launch_bounds__(32)
k_whiten(const float* __restrict__ statsA, const float* __restrict__ statsB,
         const float* __restrict__ swm, const float* __restrict__ swv,
         const float* __restrict__ weight, const float* __restrict__ bias,
         float* __restrict__ wmOut) {
  const int inst = blockIdx.x;
  const int gi = inst & 15, cb = gi * CPG;
  const int lane = threadIdx.x & 31;
  const int lane16 = lane & 15, half = lane >> 4, kh = half * 2;

  __shared__ float sMeanIn[16], sMeanBn[16], sMean[16];
  __shared__ float sCov[256], sP[256], sT[256], sW2[256];
  __shared__ float sTr;

  // softmax over the two switch weights (computed redundantly per lane)
  float a0 = swm[0], a1 = swm[1];
  float am = fmaxf(a0, a1);
  float e0 = __expf(a0 - am), e1 = __expf(a1 - am);
  float mw0 = e0 / (e0 + e1), mw1 = 1.f - mw0;
  float b0 = swv[0], b1 = swv[1];
  float bm = fmaxf(b0, b1);
  float f0 = __expf(b0 - bm), f1 = __expf(b1 - bm);
  float vw0 = f0 / (f0 + f1), vw1 = 1.f - vw0;

  const float* qin = statsA + (size_t)inst * SSTR;
  const float* qbn = statsB + (size_t)gi * SSTR;

  if (lane == 0) sTr = 0.f;
  if (lane < 16) {
    float mi = qin[256 + lane] * (1.f / (float)HWSZ);
    float mb = qbn[256 + lane] * (1.f / NHWF);
    sMeanIn[lane] = mi;
    sMeanBn[lane] = mb;
    sMean[lane] = mw0 * mb + mw1 * mi;
  }
  __syncthreads();

  // mixed covariance + eps*I ; P = I
#pragma unroll
  for (int j = 0; j < 8; ++j) {
    int e = lane * 8 + j, r = e >> 4, cidx = e & 15;
    float cin = qin[e] * (1.f / (float)HWSZ) - sMeanIn[r] * sMeanIn[cidx];
    float cbn = qbn[e] * (1.f / NHWF) - sMeanBn[r] * sMeanBn[cidx];
    sCov[e] = vw0 * cbn + vw1 * cin + ((r == cidx) ? EPSV : 0.f);
    sP[e] = (r == cidx) ? 1.f : 0.f;
  }
  __syncthreads();
  if (lane < 16) atomicAdd(&sTr, sCov[lane * 17]);   // trace
  __syncthreads();
  const float rTr = 1.f / sTr;
#pragma unroll
  for (int j = 0; j < 8; ++j) sCov[lane * 8 + j] *= rTr;   // covN
  __syncthreads();

  // Newton-Schulz: P = 1.5P - 0.5*(P*P*P)*covN, 5 iterations, all WMMA
  for (int it = 0; it < 5; ++it) {
    v8f d2 = mul16(sP, sP, lane16, kh);        // P^2
    __syncthreads();
    st16(sT, lane16, half, d2);
    __syncthreads();
    v8f d3 = mul16(sT, sP, lane16, kh);        // P^3
    __syncthreads();
    st16(sT, lane16, half, d3);
    __syncthreads();
    v8f t = mul16(sT, sCov, lane16, kh);       // P^3 * covN
    v8f np;
#pragma unroll
    for (int v = 0; v < 8; ++v)
      np[v] = 1.5f * sP[(v + half * 8) * 16 + lane16] - 0.5f * t[v];
    __syncthreads();
    st16(sP, lane16, half, np);
    __syncthreads();
  }

  // fold: wm2 = diag(weight) * P * sqrt(rTr); wbias = bias - wm2 @ mean
  const float sc = sqrtf(rTr);
#pragma unroll
  for (int j = 0; j < 8; ++j) {
    int e = lane * 8 + j, r = e >> 4;
    sW2[e] = weight[cb + r] * sP[e] * sc;
  }
  __syncthreads();

  float* dst = wmOut + (size_t)inst * SSTR;
#pragma unroll
  for (int j = 0; j < 8; ++j) {
    int e = lane * 8 + j;
    dst[e] = sW2[e];
  }
  if (lane < 16) {
    float acc = bias[cb + lane];
#pragma unroll
    for (int k = 0; k < 16; ++k) acc -= sW2[lane * 16 + k] * sMean[k];
    dst[256 + lane] = acc;
  }
}

// ---------------------------------------------------------------------------
// K3: out = wm2 @ x + wbias. One block (8 waves) per instance; each wave builds
// 16x16 output tiles with 4 chained f32 WMMAs. Output stores are NON-TEMPORAL:
// out is write-once, so NT stores keep x resident in the 192 MB L2 (populated
// by K1), turning K3's x re-reads into L2 hits (~256 MB HBM total instead of
// ~384 MB).
// ---------------------------------------------------------------------------
__global__ void k_apply(const float* __restrict__ x, const float* __restrict__ wmIn,
                        float* __restrict__ out) {
  const int inst = blockIdx.x;
  const int n = inst >> 4, gi = inst & 15, cb = gi * CPG;
  const int tid = threadIdx.x, lane = tid & 31, wave = tid >> 5;
  const int lane16 = lane & 15, half = lane >> 4, kh = half * 2;

  __shared__ float sW[256];
  __shared__ float sWb[16];
  const float* src = wmIn + (size_t)inst * SSTR;
  if (tid < 256) sW[tid] = src[tid];
  if (tid < 16) sWb[tid] = src[256 + tid];
  __syncthreads();

  // loop-invariant A fragments (wm2) and per-row bias
  v2f afr[4];
#pragma unroll
  for (int t = 0; t < 4; ++t) {
    int k = 4 * t + kh;
    afr[t].x = sW[lane16 * 16 + k];
    afr[t].y = sW[lane16 * 16 + k + 1];
  }
  float wb[8];
#pragma unroll
  for (int v = 0; v < 8; ++v) wb[v] = sWb[v + half * 8];

  const float* px = x + (size_t)(n * CCH + cb) * HWSZ;
  float* po = out + (size_t)(n * CCH + cb) * HWSZ;

  for (int tile = wave; tile < 256; tile += 8) {
    const int n0 = tile * 16 + lane16;
    v8f d = {};
#pragma unroll
    for (int t = 0; t < 4; ++t) {
      int k = 4 * t + kh;
      v2f b;
      b.x = px[(size_t)k * HWSZ + n0];         // coalesced 64B per half-wave
      b.y = px[(size_t)(k + 1) * HWSZ + n0];
      d = wmma4(afr[t], b, d);
    }
#pragma unroll
    for (int v = 0; v < 8; ++v)
      __builtin_nontemporal_store(d[v] + wb[v],
                                  po + (size_t)(v + half * 8) * HWSZ + n0);
  }
}

// ---------------------------------------------------------------------------
extern "C" void kernel_launch(void* const* d_in, const int* in_sizes, int n_in,
                              void* d_out, int out_size, void* d_ws, size_t ws_size,
                              hipStream_t stream) {
  const float* x      = (const float*)d_in[0];
  const float* swm    = (const float*)d_in[1];
  const float* swv    = (const float*)d_in[2];
  const float* weight = (const float*)d_in[3];
  const float* bias   = (const float*)d_in[4];
  float* out = (float*)d_out;

  float* wsf    = (float*)d_ws;
  float* statsA = wsf;                                          // 512*288 floats
  float* statsB = wsf + (size_t)512 * SSTR;                     // 16*288 floats
  float* wmBuf  = wsf + (size_t)512 * SSTR + (size_t)16 * SSTR; // 512*288 floats

  k_stats <<<512, 256, 0, stream>>>(x, statsA);
  k_bnagg <<< 16, 256, 0, stream>>>(statsA, statsB);
  k_whiten<<<512,  32, 0, stream>>>(statsA, statsB, swm, swv, weight, bias, wmBuf);
  k_apply <<<512, 256, 0, stream>>>(x, wmBuf, out);
}